// Attention2_5669356836446
// MI455X (gfx1250) — compile-verified
//
#include <hip/hip_runtime.h>
#include <hip/hip_bf16.h>
#include <stdint.h>

// Problem constants (match reference)
#define B_   128
#define S_   1024
#define RNN_ 1024
#define HID_ 512
#define MIN_VALUE_ (-100000000.0f)

typedef float v2f __attribute__((ext_vector_type(2)));
typedef float v8f __attribute__((ext_vector_type(8)));

// Hardware tanh (CDNA5 V_TANH_F32 trans op). tanhf() would be a ~20-op OCML
// sequence and make the 256MB streaming kernel VALU-bound (needs ~6.1e12
// tanh/s to keep up with 23.3 TB/s); the 1-instruction trans op fits the roof.
__device__ __forceinline__ float tanh_hw(float x) {
#if __has_builtin(__builtin_amdgcn_tanhf)
    return __builtin_amdgcn_tanhf(x);
#else
    float r;
    asm volatile("v_tanh_f32 %0, %1" : "=v"(r) : "v"(x));
    return r;
#endif
}

// ---------------------------------------------------------------------------
// Kernel 1: att_h[B,HID] = h[B,RNN] @ W[HID,RNN]^T + bias, via f32 WMMA.
// One wave per 16x16 output tile. K-loop: 256 iters of v_wmma_f32_16x16x4_f32.
//   A 16x4  f32: lane L -> M=L%16; VGPR j -> K = k0 + 2*(L/16) + j
//   B 4x16  f32: lane L -> N=L%16; VGPR j -> K = k0 + 2*(L/16) + j
//   C/D 16x16 f32: VGPR r -> M = r + 8*(L/16), N = L%16
// ---------------------------------------------------------------------------
__global__ __launch_bounds__(256) void k_h2att_wmma(
    const float* __restrict__ h, const float* __restrict__ W,
    const float* __restrict__ bias, float* __restrict__ att_h)
{
    const int lane = threadIdx.x & 31;
    const int wave = threadIdx.x >> 5;
    const int tile = blockIdx.x * 8 + wave;      // 256 tiles: 8 (M) x 32 (N)
    const int tm = (tile & 7) << 4;              // M tile base
    const int tn = (tile >> 3) << 4;             // N tile base
    const int lo = lane & 15;
    const int hi = lane >> 4;

    const float* aRow = h + (size_t)(tm + lo) * RNN_ + 2 * hi;
    const float* bRow = W + (size_t)(tn + lo) * RNN_ + 2 * hi;

    v8f c = {};
#pragma unroll 4
    for (int k = 0; k < RNN_; k += 4) {
        v2f a = *(const v2f*)(aRow + k);
        v2f b = *(const v2f*)(bRow + k);
        c = __builtin_amdgcn_wmma_f32_16x16x4_f32(
                false, a, false, b, (short)0, c, false, false);
    }

    const int n = tn + lo;
    const float bv = bias[n];
#pragma unroll
    for (int r = 0; r < 8; ++r) {
        const int m = tm + r + 8 * hi;
        att_h[(size_t)m * HID_ + n] = c[r] + bv;
    }
}

// ---------------------------------------------------------------------------
// Kernel 2: scores[b,s] = sum_d tanh(p_att[b,s,d] + att_h[b,d]) * w_alpha[d]
//                         + b_alpha ; masked -> MIN_VALUE.
// One wave per 8 consecutive s rows. att_h[b,:] and w_alpha cached in
// REGISTERS (16 values/lane) -> hot loop is: coalesced b32 load, v_add,
// v_tanh, v_fma. Streams the 256 MB p_att_feats tensor exactly once.
// ---------------------------------------------------------------------------
__global__ __launch_bounds__(256) void k_scores(
    const float* __restrict__ p_att, const float* __restrict__ att_h,
    const uint8_t* __restrict__ mask, const float* __restrict__ w_alpha,
    const float* __restrict__ b_alpha, float* __restrict__ scores)
{
    const int b    = blockIdx.y;
    const int wave = threadIdx.x >> 5;
    const int lane = threadIdx.x & 31;
    const int s0   = blockIdx.x * 64 + wave * 8;   // 8 s-rows per wave

    // Cache this wave's slice of att_h row and w_alpha in registers.
    float ah[16], wa[16];
    const float* ahp = att_h + (size_t)b * HID_;
#pragma unroll
    for (int i = 0; i < 16; ++i) {
        ah[i] = ahp[lane + 32 * i];       // served from L2 after kernel 1
        wa[i] = w_alpha[lane + 32 * i];
    }
    const float ba = b_alpha[0];

    for (int j = 0; j < 8; ++j) {
        const int s = s0 + j;
        const float* row = p_att + ((size_t)b * S_ + s) * HID_;
        float acc = 0.f;
#pragma unroll
        for (int i = 0; i < 16; ++i)
            acc = fmaf(tanh_hw(row[lane + 32 * i] + ah[i]), wa[i], acc);

#pragma unroll
        for (int off = 16; off > 0; off >>= 1)
            acc += __shfl_xor(acc, off, 32);

        if (lane == 0) {
            float sc = acc + ba;
            if (mask[(size_t)b * S_ + s]) sc = MIN_VALUE_;
            scores[(size_t)b * S_ + s] = sc;
        }
    }
}

// ---------------------------------------------------------------------------
// Kernel 3: softmax over S per batch row. One block per b, 4 elems/thread.
// ---------------------------------------------------------------------------
__global__ __launch_bounds__(256) void k_softmax(
    const float* __restrict__ scores, float* __restrict__ weight)
{
    __shared__ float red[8];
    const int b = blockIdx.x;
    const float* row = scores + (size_t)b * S_;

    float v[4];
    float m = -3.4e38f;
#pragma unroll
    for (int i = 0; i < 4; ++i) {
        v[i] = row[threadIdx.x + 256 * i];
        m = fmaxf(m, v[i]);
    }
#pragma unroll
    for (int off = 16; off > 0; off >>= 1)
        m = fmaxf(m, __shfl_xor(m, off, 32));
    if ((threadIdx.x & 31) == 0) red[threadIdx.x >> 5] = m;
    __syncthreads();
    float gmax = red[0];
#pragma unroll
    for (int i = 1; i < 8; ++i) gmax = fmaxf(gmax, red[i]);

    float sum = 0.f;
#pragma unroll
    for (int i = 0; i < 4; ++i) {
        v[i] = __expf(v[i] - gmax);
        sum += v[i];
    }
#pragma unroll
    for (int off = 16; off > 0; off >>= 1)
        sum += __shfl_xor(sum, off, 32);
    __syncthreads();                       // red reuse barrier
    if ((threadIdx.x & 31) == 0) red[threadIdx.x >> 5] = sum;
    __syncthreads();
    float tot = 0.f;
#pragma unroll
    for (int i = 0; i < 8; ++i) tot += red[i];
    const float inv = __frcp_rn(tot);

    float* out = weight + (size_t)b * S_;
#pragma unroll
    for (int i = 0; i < 4; ++i)
        out[threadIdx.x + 256 * i] = v[i] * inv;
}

// ---------------------------------------------------------------------------
// Kernel 4: att_res[b,:] = sum_s weight[b,s] * att_feats[b,s,:].
// Column-partitioned: grid (4, B); thread owns one RNN column; loops all S.
// Streams the 512 MB att_feats tensor exactly once, fully coalesced
// (wave32 x b32 = contiguous 128B per load).
// ---------------------------------------------------------------------------
__global__ __launch_bounds__(256) void k_wsum(
    const float* __restrict__ att_feats, const float* __restrict__ weight,
    float* __restrict__ out)
{
    __shared__ float w[S_];
    const int b = blockIdx.y;
    const int col = blockIdx.x * 256 + threadIdx.x;

    for (int i = threadIdx.x; i < S_; i += 256)
        w[i] = weight[(size_t)b * S_ + i];
    __syncthreads();

    const float* base = att_feats + (size_t)b * S_ * RNN_ + col;
    float acc = 0.f;
#pragma unroll 8
    for (int s = 0; s < S_; ++s) {
        if ((s & 63) == 0 && s + 64 < S_)                 // global_prefetch
            __builtin_prefetch(base + (size_t)(s + 64) * RNN_, 0, 0);
        acc = fmaf(w[s], base[(size_t)s * RNN_], acc);
    }
    out[(size_t)b * RNN_ + col] = acc;
}

// ---------------------------------------------------------------------------
// Host launcher. Inputs (setup_inputs order):
//  0 h[B,RNN] f32          1 att_feats[B,S,RNN] f32   2 p_att_feats[B,S,HID] f32
//  3 mask[B,S] bool(u8)    4 W_h2att[HID,RNN] f32     5 b_h2att[HID] f32
//  6 w_alpha[HID] f32      7 b_alpha[1] f32
// Output: att_res[B,RNN] f32.
// Workspace: att_h (256KB) | scores (512KB) | weight (512KB) = 1.25 MB.
// ---------------------------------------------------------------------------
extern "C" void kernel_launch(void* const* d_in, const int* in_sizes, int n_in,
                              void* d_out, int out_size, void* d_ws, size_t ws_size,
                              hipStream_t stream) {
    const float*   h        = (const float*)d_in[0];
    const float*   att_f    = (const float*)d_in[1];
    const float*   p_att    = (const float*)d_in[2];
    const uint8_t* mask     = (const uint8_t*)d_in[3];
    const float*   W_h2att  = (const float*)d_in[4];
    const float*   b_h2att  = (const float*)d_in[5];
    const float*   w_alpha  = (const float*)d_in[6];
    const float*   b_alpha  = (const float*)d_in[7];
    float*         out      = (float*)d_out;

    char* ws = (char*)d_ws;
    float* att_h  = (float*)(ws);                                   // B*HID
    float* scores = (float*)(ws + (size_t)B_ * HID_ * 4);           // B*S
    float* weight = (float*)(ws + (size_t)B_ * HID_ * 4
                                + (size_t)B_ * S_  * 4);            // B*S

    // 1) WMMA GEMM: 256 tiles of 16x16, 8 waves/block -> 32 blocks.
    k_h2att_wmma<<<dim3(32), dim3(256), 0, stream>>>(h, W_h2att, b_h2att, att_h);

    // 2) Fused tanh+dot+mask -> scores. grid (S/64, B).
    k_scores<<<dim3(S_ / 64, B_), dim3(256), 0, stream>>>(
        p_att, att_h, mask, w_alpha, b_alpha, scores);

    // 3) Softmax per batch row.
    k_softmax<<<dim3(B_), dim3(256), 0, stream>>>(scores, weight);

    // 4) Weighted sum over S. grid (RNN/256, B).
    k_wsum<<<dim3(RNN_ / 256, B_), dim3(256), 0, stream>>>(att_f, weight, out);
}